// AttnDecoderWithMemory_38620345926140
// MI455X (gfx1250) — compile-verified
//
#include <hip/hip_runtime.h>
#include <math.h>

// ---- problem dimensions (fixed by reference) ----
#define B_  64
#define H_  512
#define V_  50000
#define L_  400
#define M_  128
#define D_  64
#define C_  64
#define HC_ 260          // 2*D + M + 4
#define EPS_ 1e-8f

typedef __attribute__((ext_vector_type(2))) float v2f;
typedef __attribute__((ext_vector_type(8))) float v8f;

__device__ __forceinline__ float sigmoidf_(float x) { return 1.0f / (1.0f + expf(-x)); }

// ============================================================================
// Strip GEMM (N large): C[64 x N] (+)= A[64 x K] @ W[N x K]^T (+ bias[N])
// One wave owns a 16-wide N strip across ALL 64 output rows (4 accumulators),
// so each B (weight) fragment loaded from the HBM stream feeds 4 WMMAs.
// V_WMMA_F32_16X16X4_F32, K stepped by 4.  Requires N%16==0, K%4==0.
// ============================================================================
__global__ void gemm_wmma_f32_strip(const float* __restrict__ A, const float* __restrict__ W,
                                    const float* __restrict__ bias, float* __restrict__ C,
                                    int N, int K, int accumulate)
{
    const int lane = threadIdx.x & 31;
    const int wave = threadIdx.x >> 5;
    const int ntn  = N >> 4;
    const long strip = (long)blockIdx.x * (blockDim.x >> 5) + wave;
    if (strip >= ntn) return;                 // wave-uniform; EXEC stays all-ones

    const int li   = lane & 15;
    const int hi   = lane >> 4;               // 0: lanes 0-15, 1: lanes 16-31
    const int colB = (int)strip * 16 + li;    // output column == W row
    const int koff = hi * 2;                  // K sub-offset within group of 4

    v8f acc[4];
    const float bval = bias ? bias[colB] : 0.0f;
    if (accumulate) {
        #pragma unroll
        for (int tm = 0; tm < 4; ++tm)
            #pragma unroll
            for (int r = 0; r < 8; ++r) {
                int m = tm * 16 + r + hi * 8;
                acc[tm][r] = C[(size_t)m * N + colB] + bval;
            }
    } else {
        #pragma unroll
        for (int tm = 0; tm < 4; ++tm)
            #pragma unroll
            for (int r = 0; r < 8; ++r) acc[tm][r] = bval;
    }

    const float* Wrow = W + (size_t)colB * K;
    const float* A0   = A + (size_t)(li +  0) * K;   // A rows for the 4 M-tiles
    const float* A1   = A + (size_t)(li + 16) * K;
    const float* A2   = A + (size_t)(li + 32) * K;
    const float* A3   = A + (size_t)(li + 48) * K;

    for (int k0 = 0; k0 < K; k0 += 4) {
        if (k0 + 128 < K)                               // keep the HBM stream fed
            __builtin_prefetch(Wrow + k0 + 128, 0, 0);  // -> global_prefetch_b8
        v2f b, a0, a1, a2, a3;
        b.x  = Wrow[k0 + koff]; b.y  = Wrow[k0 + koff + 1];
        a0.x = A0[k0 + koff];   a0.y = A0[k0 + koff + 1];
        a1.x = A1[k0 + koff];   a1.y = A1[k0 + koff + 1];
        a2.x = A2[k0 + koff];   a2.y = A2[k0 + koff + 1];
        a3.x = A3[k0 + koff];   a3.y = A3[k0 + koff + 1];
        acc[0] = __builtin_amdgcn_wmma_f32_16x16x4_f32(false, a0, false, b, (short)0, acc[0], false, false);
        acc[1] = __builtin_amdgcn_wmma_f32_16x16x4_f32(false, a1, false, b, (short)0, acc[1], false, false);
        acc[2] = __builtin_amdgcn_wmma_f32_16x16x4_f32(false, a2, false, b, (short)0, acc[2], false, false);
        acc[3] = __builtin_amdgcn_wmma_f32_16x16x4_f32(false, a3, false, b, (short)0, acc[3], false, false);
    }

    #pragma unroll
    for (int tm = 0; tm < 4; ++tm)
        #pragma unroll
        for (int r = 0; r < 8; ++r) {
            int m = tm * 16 + r + hi * 8;
            C[(size_t)m * N + colB] = acc[tm][r];
        }
}

// ============================================================================
// Tile GEMM (N small): one 16x16 tile per wave (keeps parallelism when N=64).
// ============================================================================
__global__ void gemm_wmma_f32_tile(const float* __restrict__ A, const float* __restrict__ W,
                                   const float* __restrict__ bias, float* __restrict__ C,
                                   int N, int K, int accumulate)
{
    const int lane = threadIdx.x & 31;
    const int wave = threadIdx.x >> 5;
    const int ntm  = B_ >> 4;
    const int ntn  = N >> 4;
    const long tile = (long)blockIdx.x * (blockDim.x >> 5) + wave;
    if (tile >= (long)ntm * ntn) return;

    const int tm   = (int)(tile % ntm);
    const int tn   = (int)(tile / ntm);
    const int li   = lane & 15;
    const int hi   = lane >> 4;
    const int rowA = tm * 16 + li;
    const int colB = tn * 16 + li;
    const int koff = hi * 2;

    v8f acc;
    const float bval = bias ? bias[colB] : 0.0f;
    if (accumulate) {
        #pragma unroll
        for (int r = 0; r < 8; ++r) {
            int m = tm * 16 + r + hi * 8;
            acc[r] = C[(size_t)m * N + colB] + bval;
        }
    } else {
        #pragma unroll
        for (int r = 0; r < 8; ++r) acc[r] = bval;
    }

    const float* Arow = A + (size_t)rowA * K;
    const float* Wrow = W + (size_t)colB * K;
    for (int k0 = 0; k0 < K; k0 += 4) {
        if (k0 + 128 < K) {
            __builtin_prefetch(Wrow + k0 + 128, 0, 0);
            __builtin_prefetch(Arow + k0 + 128, 0, 0);
        }
        v2f a, b;
        a.x = Arow[k0 + koff];
        a.y = Arow[k0 + koff + 1];
        b.x = Wrow[k0 + koff];
        b.y = Wrow[k0 + koff + 1];
        acc = __builtin_amdgcn_wmma_f32_16x16x4_f32(false, a, false, b,
                                                    (short)0, acc, false, false);
    }

    #pragma unroll
    for (int r = 0; r < 8; ++r) {
        int m = tm * 16 + r + hi * 8;
        C[(size_t)m * N + colB] = acc[r];
    }
}

// ============================================================================
// Elementwise / reduction kernels
// ============================================================================
__global__ void embed_gather(const int* __restrict__ ids, const float* __restrict__ emb,
                             float* __restrict__ out)
{
    int i = blockIdx.x * blockDim.x + threadIdx.x;
    if (i >= B_ * H_) return;
    int b = i / H_, h = i % H_;
    out[i] = emb[(size_t)ids[b] * H_ + h];
}

__global__ void concat2(const float* __restrict__ a, int na,
                        const float* __restrict__ b, int nb,
                        float* __restrict__ dst, int rows)
{
    int n = na + nb;
    int i = blockIdx.x * blockDim.x + threadIdx.x;
    if (i >= rows * n) return;
    int r = i / n, c = i % n;
    dst[i] = (c < na) ? a[(size_t)r * na + c] : b[(size_t)r * nb + (c - na)];
}

__global__ void add2(const float* __restrict__ a, const float* __restrict__ b,
                     float* __restrict__ dst, int n)
{
    int i = blockIdx.x * blockDim.x + threadIdx.x;
    if (i < n) dst[i] = a[i] + b[i];
}

// aw = softmax(tc + ia + ts) along L, one block (256 thr) per batch row
__global__ void attn_softmax(const float* __restrict__ tc, const float* __restrict__ ia,
                             const float* __restrict__ ts, float* __restrict__ aw)
{
    __shared__ float sm[256];
    int b = blockIdx.x, t = threadIdx.x;
    float lmax = -3.4e38f;
    for (int l = t; l < L_; l += 256) {
        float s = tc[b * L_ + l] + ia[b * L_ + l] + ts[b * L_ + l];
        aw[b * L_ + l] = s;
        lmax = fmaxf(lmax, s);
    }
    sm[t] = lmax; __syncthreads();
    for (int s = 128; s > 0; s >>= 1) { if (t < s) sm[t] = fmaxf(sm[t], sm[t + s]); __syncthreads(); }
    float mx = sm[0]; __syncthreads();
    float lsum = 0.0f;
    for (int l = t; l < L_; l += 256) {
        float e = expf(aw[b * L_ + l] - mx);
        aw[b * L_ + l] = e;
        lsum += e;
    }
    sm[t] = lsum; __syncthreads();
    for (int s = 128; s > 0; s >>= 1) { if (t < s) sm[t] += sm[t + s]; __syncthreads(); }
    float inv = 1.0f / sm[0];
    for (int l = t; l < L_; l += 256) aw[b * L_ + l] *= inv;
}

// attn_applied[b,h] = sum_l aw[b,l] * eo[b,l,h]   (streams 52 MB of eo)
__global__ void attn_apply(const float* __restrict__ aw, const float* __restrict__ eo,
                           float* __restrict__ out)
{
    int b = blockIdx.x;
    for (int h = threadIdx.x; h < H_; h += blockDim.x) {
        float acc = 0.0f;
        const float* eob = eo + (size_t)b * L_ * H_ + h;
        for (int l = 0; l < L_; ++l) {
            if (l + 4 < L_)
                __builtin_prefetch(eob + (size_t)(l + 4) * H_, 0, 0);
            acc += aw[b * L_ + l] * eob[(size_t)l * H_];
        }
        out[b * H_ + h] = acc;
    }
}

// LSTM gates: g[B,4N] (i|f|g|o), cprev[B,N] -> h[B,N], c[B,N]
__global__ void lstm_elem(const float* __restrict__ g, const float* __restrict__ cprev,
                          float* __restrict__ hout, float* __restrict__ cout, int N)
{
    int i = blockIdx.x * blockDim.x + threadIdx.x;
    if (i >= B_ * N) return;
    int b = i / N, j = i % N;
    const float* gb = g + (size_t)b * 4 * N;
    float ig = sigmoidf_(gb[j]);
    float fg = sigmoidf_(gb[N + j]);
    float gg = tanhf(gb[2 * N + j]);
    float og = sigmoidf_(gb[3 * N + j]);
    float c2 = fg * cprev[i] + ig * gg;
    cout[i] = c2;
    hout[i] = og * tanhf(c2);
}

// per-(b,m) memory row sum and L2 norm
__global__ void mem_stats(const float* __restrict__ mem, float* __restrict__ rowsum,
                          float* __restrict__ rownorm)
{
    int i = blockIdx.x * blockDim.x + threadIdx.x;
    if (i >= B_ * M_) return;
    const float* p = mem + (size_t)i * D_;
    float s = 0.0f, sq = 0.0f;
    for (int d = 0; d < D_; ++d) { float v = p[d]; s += v; sq += v * v; }
    rowsum[i] = s;
    rownorm[i] = sqrtf(sq);
}

// content addressing + gating; one block (M_=128 threads) per batch row.
// heads points to read/write_heads[0,0,:] (reference broadcasts heads[0]).
__global__ void addressing(const float* __restrict__ hc, const float* __restrict__ heads,
                           const float* __restrict__ rowsum, const float* __restrict__ rownorm,
                           float* __restrict__ wout)
{
    __shared__ float sm[M_];
    int b = blockIdx.x, m = threadIdx.x;
    const float* h = hc + (size_t)b * HC_;
    float key  = h[m];
    float kstr = expf(h[D_]);
    float gate = sigmoidf_(h[D_ + 1]);
    float num  = key * rowsum[b * M_ + m];
    float nk   = fmaxf(fabsf(key) * 8.0f, EPS_);      // sqrt(D)=8
    float nm   = fmaxf(rownorm[b * M_ + m], EPS_);
    float s    = kstr * (num / (nk * nm));
    sm[m] = s; __syncthreads();
    for (int st = 64; st > 0; st >>= 1) { if (m < st) sm[m] = fmaxf(sm[m], sm[m + st]); __syncthreads(); }
    float mx = sm[0]; __syncthreads();
    float e = expf(s - mx);
    sm[m] = e; __syncthreads();
    for (int st = 64; st > 0; st >>= 1) { if (m < st) sm[m] += sm[m + st]; __syncthreads(); }
    float content = e / sm[0];
    wout[b * M_ + m] = gate * content + (1.0f - gate) * heads[m];
}

// read_in[b,d] = sum_m rw[b,m] * mem[b,m,d]
__global__ void read_reduce(const float* __restrict__ rw, const float* __restrict__ mem,
                            float* __restrict__ out)
{
    int i = blockIdx.x * blockDim.x + threadIdx.x;
    if (i >= B_ * D_) return;
    int b = i / D_, d = i % D_;
    float acc = 0.0f;
    for (int m = 0; m < M_; ++m)
        acc += rw[b * M_ + m] * mem[((size_t)b * M_ + m) * D_ + d];
    out[i] = acc;
}

// new_mem = mem * (1 - ww*erase) + ww*add   (erase/add derived from write hidden)
__global__ void memory_update(const float* __restrict__ mem, const float* __restrict__ ww,
                              const float* __restrict__ wh, float* __restrict__ out)
{
    int i = blockIdx.x * blockDim.x + threadIdx.x;
    if (i >= B_ * M_ * D_) return;
    int b = i / (M_ * D_);
    int m = (i % (M_ * D_)) / D_;
    const float* h = wh + (size_t)b * HC_;
    float we = sigmoidf_(h[D_ + 4 + m]);
    float wa = sigmoidf_(h[2 * D_ + 4 + m]);
    float w0 = ww[b * M_ + m];
    out[i] = mem[i] * (1.0f - w0 * we) + w0 * wa;
}

// in-place log_softmax over V per row; one block (256 thr) per batch row
__global__ void log_softmax_rows(float* __restrict__ x)
{
    __shared__ float sm[256];
    int b = blockIdx.x, t = threadIdx.x;
    float* row = x + (size_t)b * V_;
    float lmax = -3.4e38f;
    for (int v = t; v < V_; v += 256) lmax = fmaxf(lmax, row[v]);
    sm[t] = lmax; __syncthreads();
    for (int s = 128; s > 0; s >>= 1) { if (t < s) sm[t] = fmaxf(sm[t], sm[t + s]); __syncthreads(); }
    float mx = sm[0]; __syncthreads();
    float lsum = 0.0f;
    for (int v = t; v < V_; v += 256) lsum += expf(row[v] - mx);
    sm[t] = lsum; __syncthreads();
    for (int s = 128; s > 0; s >>= 1) { if (t < s) sm[t] += sm[t + s]; __syncthreads(); }
    float lse = logf(sm[0]) + mx;
    for (int v = t; v < V_; v += 256) row[v] -= lse;
}

// ============================================================================
extern "C" void kernel_launch(void* const* d_in, const int* in_sizes, int n_in,
                              void* d_out, int out_size, void* d_ws, size_t ws_size,
                              hipStream_t stream)
{
    const int*   input_ids  = (const int*)  d_in[0];
    const float* h0         = (const float*)d_in[1];
    const float* c0         = (const float*)d_in[2];
    const float* eo         = (const float*)d_in[3];
    const float* coverage   = (const float*)d_in[5];
    const float* memory     = (const float*)d_in[6];
    const float* read_heads = (const float*)d_in[7];
    const float* write_heads= (const float*)d_in[8];
    const float* read_h     = (const float*)d_in[9];
    const float* read_c     = (const float*)d_in[10];
    const float* write_h    = (const float*)d_in[11];
    const float* write_c    = (const float*)d_in[12];
    const float* emb        = (const float*)d_in[13];
    const float* attn_W     = (const float*)d_in[14];
    const float* attn_b     = (const float*)d_in[15];
    const float* cov_W      = (const float*)d_in[16];
    const float* state_W    = (const float*)d_in[17];
    const float* comb_W     = (const float*)d_in[18];
    const float* comb_b     = (const float*)d_in[19];
    const float* rpre_W     = (const float*)d_in[20];
    const float* rpre_b     = (const float*)d_in[21];
    const float* wpre_W     = (const float*)d_in[22];
    const float* wpre_b     = (const float*)d_in[23];
    const float* r_Wih      = (const float*)d_in[24];
    const float* r_Whh      = (const float*)d_in[25];
    const float* r_bih      = (const float*)d_in[26];
    const float* r_bhh      = (const float*)d_in[27];
    const float* w_Wih      = (const float*)d_in[28];
    const float* w_Whh      = (const float*)d_in[29];
    const float* w_bih      = (const float*)d_in[30];
    const float* w_bhh      = (const float*)d_in[31];
    const float* l_Wih      = (const float*)d_in[32];
    const float* l_Whh      = (const float*)d_in[33];
    const float* l_bih      = (const float*)d_in[34];
    const float* l_bhh      = (const float*)d_in[35];
    const float* out_W      = (const float*)d_in[36];
    const float* out_b      = (const float*)d_in[37];

    // output slots (flat, return order)
    float* out     = (float*)d_out;
    float* logits  = out;                             // B*V   (log_probs)
    float* h1      = logits + (size_t)B_ * V_;        // B*H
    float* c1      = h1 + B_ * H_;                    // B*H
    float* new_mem = c1 + B_ * H_;                    // B*M*D
    float* new_cov = new_mem + (size_t)B_ * M_ * D_;  // B*L

    // workspace allocator
    float* ws = (float*)d_ws;
    size_t off = 0;
    auto alloc = [&](size_t n) { float* p = ws + off; off += n; return p; };

    float* embedded = alloc(B_ * H_);
    float* cat_ih   = alloc(B_ * 2 * H_);
    float* ia       = alloc(B_ * L_);
    float* tc       = alloc(B_ * L_);
    float* ts       = alloc(B_ * L_);
    float* aw       = alloc(B_ * L_);
    float* attn_ap  = alloc(B_ * H_);
    float* cat_comb = alloc(B_ * 2 * H_);
    float* out0     = alloc(B_ * H_);
    float* hm       = alloc((size_t)B_ * (H_ + M_ * D_));
    float* rpre_x   = alloc(B_ * C_);
    float* wpre_x   = alloc(B_ * C_);
    float* g_r      = alloc(B_ * 4 * HC_);
    float* g_w      = alloc(B_ * 4 * HC_);
    float* rh       = alloc(B_ * HC_);
    float* rc2      = alloc(B_ * HC_);
    float* wh       = alloc(B_ * HC_);
    float* wc2      = alloc(B_ * HC_);
    float* rowsum   = alloc(B_ * M_);
    float* rownorm  = alloc(B_ * M_);
    float* rw       = alloc(B_ * M_);
    float* wwgt     = alloc(B_ * M_);
    float* read_in  = alloc(B_ * D_);
    float* cat_l    = alloc(B_ * (H_ + D_));
    float* g_l      = alloc(B_ * 4 * H_);

    auto gemm = [&](const float* A, const float* W, const float* bias, float* Cm,
                    int N, int K, int accFlag) {
        if (N >= 256) {                                // strip kernel: B-frag reuse x4
            int strips = N / 16;
            int blocks = (strips + 7) / 8;             // 8 waves / 256-thread block
            gemm_wmma_f32_strip<<<blocks, 256, 0, stream>>>(A, W, bias, Cm, N, K, accFlag);
        } else {                                       // tile kernel: keep wave count up
            long tiles = (long)(B_ / 16) * (N / 16);
            int blocks = (int)((tiles + 7) / 8);
            gemm_wmma_f32_tile<<<blocks, 256, 0, stream>>>(A, W, bias, Cm, N, K, accFlag);
        }
    };
    auto ew = [&](int n) { return dim3((n + 255) / 256); };

    // 1) embedding + attention input
    embed_gather<<<ew(B_ * H_), 256, 0, stream>>>(input_ids, emb, embedded);
    concat2<<<ew(B_ * 2 * H_), 256, 0, stream>>>(embedded, H_, h0, H_, cat_ih, B_);
    gemm(cat_ih, attn_W, attn_b, ia, L_, 2 * H_, 0);                   // ia

    // 2) new_coverage = coverage + ia (also the GEMM input for tc)
    add2<<<ew(B_ * L_), 256, 0, stream>>>(coverage, ia, new_cov, B_ * L_);
    gemm(new_cov, cov_W, nullptr, tc, L_, L_, 0);                      // tc
    gemm(c0, state_W, nullptr, ts, L_, H_, 0);                         // ts

    // 3) attention weights + applied context
    attn_softmax<<<B_, 256, 0, stream>>>(tc, ia, ts, aw);
    attn_apply<<<B_, 256, 0, stream>>>(aw, eo, attn_ap);
    concat2<<<ew(B_ * 2 * H_), 256, 0, stream>>>(embedded, H_, attn_ap, H_, cat_comb, B_);
    gemm(cat_comb, comb_W, comb_b, out0, H_, 2 * H_, 0);               // out0

    // 4) head pre-projections from [h0 | memory]
    concat2<<<ew((int)((size_t)B_ * (H_ + M_ * D_))), 256, 0, stream>>>(
        h0, H_, memory, M_ * D_, hm, B_);
    gemm(hm, rpre_W, rpre_b, rpre_x, C_, H_ + M_ * D_, 0);
    gemm(hm, wpre_W, wpre_b, wpre_x, C_, H_ + M_ * D_, 0);

    // 5) read / write controller LSTMs
    gemm(rpre_x, r_Wih, r_bih, g_r, 4 * HC_, C_, 0);
    gemm(read_h, r_Whh, r_bhh, g_r, 4 * HC_, HC_, 1);
    lstm_elem<<<ew(B_ * HC_), 256, 0, stream>>>(g_r, read_c, rh, rc2, HC_);
    gemm(wpre_x, w_Wih, w_bih, g_w, 4 * HC_, C_, 0);
    gemm(write_h, w_Whh, w_bhh, g_w, 4 * HC_, HC_, 1);
    lstm_elem<<<ew(B_ * HC_), 256, 0, stream>>>(g_w, write_c, wh, wc2, HC_);

    // 6) content addressing
    mem_stats<<<ew(B_ * M_), 256, 0, stream>>>(memory, rowsum, rownorm);
    addressing<<<B_, M_, 0, stream>>>(rh, read_heads, rowsum, rownorm, rw);
    addressing<<<B_, M_, 0, stream>>>(wh, write_heads, rowsum, rownorm, wwgt);

    // 7) read vector + main LSTM
    read_reduce<<<ew(B_ * D_), 256, 0, stream>>>(rw, memory, read_in);
    concat2<<<ew(B_ * (H_ + D_)), 256, 0, stream>>>(out0, H_, read_in, D_, cat_l, B_);
    gemm(cat_l, l_Wih, l_bih, g_l, 4 * H_, H_ + D_, 0);
    gemm(h0, l_Whh, l_bhh, g_l, 4 * H_, H_, 1);
    lstm_elem<<<ew(B_ * H_), 256, 0, stream>>>(g_l, c0, h1, c1, H_);

    // 8) memory write
    memory_update<<<ew(B_ * M_ * D_), 256, 0, stream>>>(memory, wwgt, wh, new_mem);

    // 9) output projection (dominant GEMM: 64 x 50000 x 512) + log_softmax
    gemm(c1, out_W, out_b, logits, V_, H_, 0);
    log_softmax_rows<<<B_, 256, 0, stream>>>(logits);
}